// ViewTransform_80204219286160
// MI455X (gfx1250) — compile-verified
//
#include <hip/hip_runtime.h>

#define B_  8
#define C_  64
#define H_  192
#define W_  640
#define HW_ (H_ * W_)

typedef float v2f __attribute__((ext_vector_type(2)));
typedef float v8f __attribute__((ext_vector_type(8)));

// ---------------------------------------------------------------------------
// Kernel 1: per-batch transform fold  A3 = (K @ T_b)[:, :3] @ K_inv ,
//           c3 = (K @ T_b)[:, 3]   via V_WMMA_F32_16X16X4_F32.
// One wave (32 lanes) per batch; 8 waves in one 256-thread block.
// A-operand (16x4, f32): lane L<16 holds A[L][0],A[L][1]; lane L>=16 holds
// A[L-16][2],A[L-16][3].  B-operand (4x16): v0 = rows 0/2, v1 = rows 1/3,
// columns striped across lane groups.  D: VGPR r = row r (lanes 0-15).
// ---------------------------------------------------------------------------
__global__ void __launch_bounds__(256)
precompute_transform_kernel(const float* __restrict__ T,
                            const float* __restrict__ K,
                            const float* __restrict__ Kinv,
                            float* __restrict__ ws) {
    const int lane = threadIdx.x & 31;
    const int b    = threadIdx.x >> 5;          // 8 waves -> 8 batches
    const float* Tb = T + b * 12;               // T_b is 3x4 row-major
    const int  row = lane & 15;
    const bool hi  = lane >= 16;

    // A[m][k] = M[m][k] = (K @ T_b)[m][k]  for k < 3;  A[m][3] = 0 (pad)
    float ax = 0.0f, ay = 0.0f;
    if (row < 3) {
        const int c0 = hi ? 2 : 0;
        ax = K[row*3+0]*Tb[0*4+c0] + K[row*3+1]*Tb[1*4+c0] + K[row*3+2]*Tb[2*4+c0];
        if (!hi) {  // k = 1
            ay = K[row*3+0]*Tb[0*4+1] + K[row*3+1]*Tb[1*4+1] + K[row*3+2]*Tb[2*4+1];
        }           // hi lanes: k = 3 -> stays 0 (pad column)
    }

    // B[k][n] = Kinv[k][n] for k<3, n<3; zero-padded elsewhere
    const int n = lane & 15;
    float bx = 0.0f, by = 0.0f;
    if (n < 3) {
        bx = hi ? Kinv[2*3 + n] : Kinv[0*3 + n];  // rows 2 / 0
        if (!hi) by = Kinv[1*3 + n];              // row 1 (hi: row 3 = pad)
    }

    v2f a;  a.x = ax; a.y = ay;
    v2f bb; bb.x = bx; bb.y = by;
    v8f acc = {};
    // D = A(16x4) x B(4x16) + 0   ->  v_wmma_f32_16x16x4_f32
    acc = __builtin_amdgcn_wmma_f32_16x16x4_f32(false, a, false, bb,
                                                (short)0, acc, false, false);

    // D[r][n], r<3, n<3 lives in VGPR r of lane n. Lane n writes column n.
    if (lane < 3) {
        float* dst = ws + b * 12;
        dst[0*3 + lane] = acc[0];
        dst[1*3 + lane] = acc[1];
        dst[2*3 + lane] = acc[2];
        // c3[lane] = (K @ T_b)[lane][3]
        dst[9 + lane] = K[lane*3+0]*Tb[3] + K[lane*3+1]*Tb[7] + K[lane*3+2]*Tb[11];
    }
}

// ---------------------------------------------------------------------------
// Kernel 2: per-pixel projection + 64-channel bilinear gather.
// Bandwidth-bound: ~252 MB img reads (L2-cached, 192 MB L2) + 252 MB
// non-temporal output stores. grid = (HW/256, B) so `b` is block-uniform
// and the 12 transform floats become scalar loads.
// ---------------------------------------------------------------------------
__global__ void __launch_bounds__(256)
warp_bilinear_kernel(const float* __restrict__ img,
                     const float* __restrict__ depth,
                     const float* __restrict__ M,
                     float* __restrict__ out) {
    const int b = blockIdx.y;
    const int n = blockIdx.x * 256 + threadIdx.x;   // pixel index, exact grid

    const float* Mb = M + b * 12;
    const float a00 = Mb[0], a01 = Mb[1], a02 = Mb[2];
    const float a10 = Mb[3], a11 = Mb[4], a12 = Mb[5];
    const float a20 = Mb[6], a21 = Mb[7], a22 = Mb[8];
    const float c0  = Mb[9], c1  = Mb[10], c2 = Mb[11];

    const float u = (float)(n % W_);
    const float v = (float)(n / W_);
    const float d = depth[b * HW_ + n];

    // p = d * (A3 @ [u,v,1]) + c3
    const float px = fmaf(d, fmaf(a00, u, fmaf(a01, v, a02)), c0);
    const float py = fmaf(d, fmaf(a10, u, fmaf(a11, v, a12)), c1);
    const float pz = fmaf(d, fmaf(a20, u, fmaf(a21, v, a22)), c2);

    const float inv = 1.0f / (pz + 1e-16f);
    const float uu = px * inv;
    const float vv = py * inv;

    const float fu = floorf(uu);
    const float fv = floorf(vv);
    const float u0 = fminf(fmaxf(fu,        0.0f), (float)(W_ - 1));
    const float u1 = fminf(fmaxf(fu + 1.0f, 0.0f), (float)(W_ - 1));
    const float v0 = fminf(fmaxf(fv,        0.0f), (float)(H_ - 1));
    const float v1 = fminf(fmaxf(fv + 1.0f, 0.0f), (float)(H_ - 1));
    const int iu0 = (int)u0, iu1 = (int)u1;
    const int iv0 = (int)v0, iv1 = (int)v1;

    const float waa = (v1 - vv) * (u1 - uu);   // (v0,u0)
    const float wab = (vv - v0) * (u1 - uu);   // (v1,u0)
    const float wac = (v1 - vv) * (uu - u0);   // (v0,u1)
    const float wad = (vv - v0) * (uu - u0);   // (v1,u1)

    const int i00 = iv0 * W_ + iu0;
    const int i10 = iv1 * W_ + iu0;
    const int i01 = iv0 * W_ + iu1;
    const int i11 = iv1 * W_ + iu1;

    const float* __restrict__ src = img + (size_t)b * (C_ * HW_);
    float*       __restrict__ dst = out + (size_t)b * (C_ * HW_) + n;

#pragma unroll 8
    for (int c = 0; c < C_; ++c) {
        const float pa = src[i00];
        const float pb = src[i10];
        const float pc = src[i01];
        const float pd = src[i11];
        float o = waa * pa;
        o = fmaf(wab, pb, o);
        o = fmaf(wac, pc, o);
        o = fmaf(wad, pd, o);
        __builtin_nontemporal_store(o, dst);   // keep L2 for img gathers
        src += HW_;
        dst += HW_;
    }
}

// ---------------------------------------------------------------------------
extern "C" void kernel_launch(void* const* d_in, const int* in_sizes, int n_in,
                              void* d_out, int out_size, void* d_ws, size_t ws_size,
                              hipStream_t stream) {
    const float* img   = (const float*)d_in[0];   // (B, C, H, W)
    const float* depth = (const float*)d_in[1];   // (B, 1, H, W)
    const float* T     = (const float*)d_in[2];   // (B, 3, 4)
    const float* K     = (const float*)d_in[3];   // (3, 3)
    const float* Kinv  = (const float*)d_in[4];   // (3, 3)
    float* out = (float*)d_out;
    float* ws  = (float*)d_ws;                    // 8 * 12 floats = 384 B

    precompute_transform_kernel<<<1, 256, 0, stream>>>(T, K, Kinv, ws);

    dim3 grid(HW_ / 256, B_);                     // 480 x 8 blocks, exact
    warp_bilinear_kernel<<<grid, 256, 0, stream>>>(img, depth, ws, out);
}